// MyModel_81587198755150
// MI455X (gfx1250) — compile-verified
//
#include <hip/hip_runtime.h>

// Fused diag-Hessian Adam update, pure streaming kernel.
// N = 33,554,432 fp32; 32 B traffic/element -> HBM-bound (~46 us @ 23.3 TB/s).

typedef float f32x4 __attribute__((ext_vector_type(4)));
typedef int   v4i   __attribute__((vector_size(16)));   // matches builtin param type

#define TPB 256          // 8 waves (wave32)
#define MAXBLK 8192

#define LR_C   7.5e-05f
#define EPS_C  1e-07f
#define CH_C   0.0405f   // 5 * 0.3^4
#define CO_C   0.54f     // 5*4 * 0.3^3

// ---- feature detection: gfx1250 async global->LDS path -------------------
#if defined(__has_builtin)
#  if __has_builtin(__builtin_amdgcn_global_load_async_to_lds_b128)
#    define USE_ASYNC 1
#  endif
#endif
#ifndef USE_ASYNC
#  define USE_ASYNC 0
#endif

__device__ __forceinline__ void wait_async_5() {
#if defined(__has_builtin) && __has_builtin(__builtin_amdgcn_s_wait_asynccnt)
    __builtin_amdgcn_s_wait_asynccnt(5);
#elif defined(__HIP_DEVICE_COMPILE__)
    asm volatile("s_wait_asynccnt 5" ::: "memory");
#endif
}
__device__ __forceinline__ void wait_async_0() {
#if defined(__has_builtin) && __has_builtin(__builtin_amdgcn_s_wait_asynccnt)
    __builtin_amdgcn_s_wait_asynccnt(0);
#elif defined(__HIP_DEVICE_COMPILE__)
    asm volatile("s_wait_asynccnt 0" ::: "memory");
#endif
}

#if USE_ASYNC
typedef __attribute__((address_space(1))) v4i* gptr_v4i;
typedef __attribute__((address_space(3))) v4i* lptr_v4i;

__device__ __forceinline__ void async_ld16(const f32x4* src, f32x4* lds_slot) {
    __builtin_amdgcn_global_load_async_to_lds_b128(
        (gptr_v4i)(src),
        (lptr_v4i)(lds_slot),
        /*imm offset*/0, /*cpol*/0);
}
#endif

// ---- scalar update (shared by vector body and tail) ----------------------
__device__ __forceinline__ float upd1(float w, float g, float h, float m, float v,
                                      float lr_bc1, float inv_bc2,
                                      float& mo, float& vo) {
    float g2   = g * g;
    float newH = fabsf(CO_C * g2 + CH_C * h) + 0.01f;
    float d2W  = g * CH_C / newH;
    mo = 0.9f * m + 0.1f * d2W;
    vo = 0.999f * v + 0.001f * g2;
    return w - lr_bc1 * mo / (sqrtf(vo * inv_bc2) + EPS_C);
}

__device__ __forceinline__ void upd4(f32x4 w, f32x4 g, f32x4 h, f32x4 m, f32x4 v,
                                     float lr_bc1, float inv_bc2,
                                     f32x4& wo, f32x4& mo, f32x4& vo) {
#pragma unroll
    for (int k = 0; k < 4; ++k) {
        float mk, vk;
        wo[k] = upd1(w[k], g[k], h[k], m[k], v[k], lr_bc1, inv_bc2, mk, vk);
        mo[k] = mk;
        vo[k] = vk;
    }
}

__global__ __launch_bounds__(TPB) void fused_hess_adam(
    const float* __restrict__ var, const float* __restrict__ g,
    const float* __restrict__ h,   const float* __restrict__ m,
    const float* __restrict__ v,   const int* __restrict__ step_p,
    float* __restrict__ var_out, float* __restrict__ m_out,
    float* __restrict__ v_out, int n)
{
    // step is a device scalar -> bias corrections computed once per thread
    const float stepf   = (float)step_p[0];
    const float lr_bc1  = LR_C / (1.0f - powf(0.9f,   stepf));
    const float inv_bc2 = 1.0f / (1.0f - powf(0.999f, stepf));

    const int tid    = threadIdx.x;
    const int n4     = n >> 2;
    const int ntiles = n4 / TPB;     // full 256-wide f32x4 tiles
    const int gs     = gridDim.x;

    const f32x4* w4 = (const f32x4*)var;
    const f32x4* g4 = (const f32x4*)g;
    const f32x4* h4 = (const f32x4*)h;
    const f32x4* m4 = (const f32x4*)m;
    const f32x4* v4 = (const f32x4*)v;
    f32x4* wo4 = (f32x4*)var_out;
    f32x4* mo4 = (f32x4*)m_out;
    f32x4* vo4 = (f32x4*)v_out;

#if USE_ASYNC
    // Double-buffered async global->LDS staging of the 5 input streams.
    // 5 streams * 2 bufs * 256 * 16B = 40 KB (8 blocks/WGP fit in 320 KB).
    // Each lane reads back only its own slot -> s_wait_asynccnt orders it,
    // no workgroup barrier needed.
    __shared__ f32x4 sW[2][TPB], sG[2][TPB], sH[2][TPB], sM[2][TPB], sV[2][TPB];

    int buf = 0;
    int t = blockIdx.x;
    if (t < ntiles) {
        int i = t * TPB + tid;
        async_ld16(&w4[i], &sW[0][tid]);
        async_ld16(&g4[i], &sG[0][tid]);
        async_ld16(&h4[i], &sH[0][tid]);
        async_ld16(&m4[i], &sM[0][tid]);
        async_ld16(&v4[i], &sV[0][tid]);
    }
    for (; t < ntiles; t += gs) {
        const int nxt = t + gs;
        if (nxt < ntiles) {
            int i = nxt * TPB + tid;
            int nb = buf ^ 1;
            async_ld16(&w4[i], &sW[nb][tid]);
            async_ld16(&g4[i], &sG[nb][tid]);
            async_ld16(&h4[i], &sH[nb][tid]);
            async_ld16(&m4[i], &sM[nb][tid]);
            async_ld16(&v4[i], &sV[nb][tid]);
            wait_async_5();          // current tile's 5 loads done (in-order)
        } else {
            wait_async_0();
        }
        f32x4 wv = sW[buf][tid], gv = sG[buf][tid], hv = sH[buf][tid];
        f32x4 mv = sM[buf][tid], vv = sV[buf][tid];
        f32x4 wo, mo, vo;
        upd4(wv, gv, hv, mv, vv, lr_bc1, inv_bc2, wo, mo, vo);
        const int oi = t * TPB + tid;
        __builtin_nontemporal_store(wo, &wo4[oi]);   // single-touch: th:NT
        __builtin_nontemporal_store(mo, &mo4[oi]);
        __builtin_nontemporal_store(vo, &vo4[oi]);
        buf ^= 1;
    }
#else
    // Plain NT b128 streaming with explicit prefetch of the next tile.
    for (int t = blockIdx.x; t < ntiles; t += gs) {
        const int i = t * TPB + tid;
        const int pi = i + gs * TPB;
        if (pi < n4) {
            __builtin_prefetch(&g4[pi], 0, 0);       // global_prefetch_b8
            __builtin_prefetch(&h4[pi], 0, 0);
            __builtin_prefetch(&m4[pi], 0, 0);
            __builtin_prefetch(&v4[pi], 0, 0);
            __builtin_prefetch(&w4[pi], 0, 0);
        }
        f32x4 wv = __builtin_nontemporal_load(&w4[i]);
        f32x4 gv = __builtin_nontemporal_load(&g4[i]);
        f32x4 hv = __builtin_nontemporal_load(&h4[i]);
        f32x4 mv = __builtin_nontemporal_load(&m4[i]);
        f32x4 vv = __builtin_nontemporal_load(&v4[i]);
        f32x4 wo, mo, vo;
        upd4(wv, gv, hv, mv, vv, lr_bc1, inv_bc2, wo, mo, vo);
        __builtin_nontemporal_store(wo, &wo4[i]);
        __builtin_nontemporal_store(mo, &mo4[i]);
        __builtin_nontemporal_store(vo, &vo4[i]);
    }
#endif

    // Generic tail (none for N = 2^25, but stay correct for any N)
    if (blockIdx.x == 0) {
        for (int idx = ntiles * TPB * 4 + tid; idx < n; idx += TPB) {
            float mo, vo;
            float wo = upd1(var[idx], g[idx], h[idx], m[idx], v[idx],
                            lr_bc1, inv_bc2, mo, vo);
            var_out[idx] = wo;
            m_out[idx]   = mo;
            v_out[idx]   = vo;
        }
    }
}

extern "C" void kernel_launch(void* const* d_in, const int* in_sizes, int n_in,
                              void* d_out, int out_size, void* d_ws, size_t ws_size,
                              hipStream_t stream) {
    (void)n_in; (void)out_size; (void)d_ws; (void)ws_size;
    const float* var = (const float*)d_in[0];
    const float* g   = (const float*)d_in[1];
    const float* h   = (const float*)d_in[2];
    const float* m   = (const float*)d_in[3];
    const float* v   = (const float*)d_in[4];
    const int*   st  = (const int*)d_in[5];
    const int n = in_sizes[0];

    float* out     = (float*)d_out;
    float* var_out = out;
    float* m_out   = out + (size_t)n;
    float* v_out   = out + 2 * (size_t)n;

    int ntiles = (n >> 2) / TPB;
    int blocks = ntiles < 1 ? 1 : (ntiles > MAXBLK ? MAXBLK : ntiles);

    fused_hess_adam<<<blocks, TPB, 0, stream>>>(var, g, h, m, v, st,
                                                var_out, m_out, v_out, n);
}